// BrainGNN_74603581931801
// MI455X (gfx1250) — compile-verified
//
#include <hip/hip_runtime.h>
#include <hip/hip_bf16.h>
#include <math.h>

// ---------------- problem constants (match reference) ----------------
#define NB       32
#define N_REG    116
#define NN       3712        // N_NODES
#define DIN      256
#define EE       37120       // explicit edges
#define ETOT     40832       // edges + self loops (EE + NN)
#define NH       8           // heads
#define LN_EPS   1e-5f

typedef __attribute__((ext_vector_type(2))) float v2f;
typedef __attribute__((ext_vector_type(8))) float v8f;

// ---------------- helpers ----------------
__device__ __forceinline__ float waveReduceSum(float v) {
    #pragma unroll
    for (int off = 16; off > 0; off >>= 1) v += __shfl_down(v, off, 32);
    return v;
}

__device__ __forceinline__ float blockReduceSum(float v, float* sm) {
    const int tid = threadIdx.x, bs = blockDim.x;
    sm[tid] = v; __syncthreads();
    for (int s = bs >> 1; s > 0; s >>= 1) {
        if (tid < s) sm[tid] += sm[tid + s];
        __syncthreads();
    }
    float r = sm[0]; __syncthreads();
    return r;
}

__device__ __forceinline__ float eluf(float v) { return v > 0.f ? v : (expf(v) - 1.f); }

// monotone float <-> orderable uint (for atomicMax-based segment max)
__device__ __forceinline__ unsigned fkey(float f) {
    unsigned b = __float_as_uint(f);
    return (b & 0x80000000u) ? ~b : (b | 0x80000000u);
}
__device__ __forceinline__ float funkey(unsigned k) {
    unsigned b = (k & 0x80000000u) ? (k & 0x7FFFFFFFu) : ~k;
    return __uint_as_float(b);
}

// ---------------- WMMA fp32 GEMM: C[N,M] = A[N,K] @ W[K,M] ----------------
// One wave per 16(row) x 64(col) output macro-tile = 4 accumulators; the A
// fragment is loaded once per k-step and reused by 4 V_WMMA_F32_16X16X4_F32.
// Layout (ISA 7.12.2, 32-bit):
//   A: lane L holds A[ti*16 + (L&15)][k0 + 2*(L>>4) + {0,1}]
//   B: lane L holds W[k0 + 2*(L>>4) + {0,1}][colbase + (L&15)]
//   C: vgpr v of lane L -> C[ti*16 + v + 8*(L>>4)][colbase + (L&15)]
__global__ void wmma_gemm_f32(const float* __restrict__ A, const float* __restrict__ W,
                              float* __restrict__ C, int N, int K, int M) {
    const int wave = (blockIdx.x * blockDim.x + threadIdx.x) >> 5;
    const int lane = threadIdx.x & 31;
    const int mt = M >> 6;                        // 64-wide column macro-tiles
    const int ti = wave / mt;
    const int cb = (wave - ti * mt) << 6;         // column base
    if (ti * 16 >= N) return;                     // wave-uniform guard
    const int half = lane >> 4, r = lane & 15;

    const float* __restrict__ arow = A + (size_t)(ti * 16 + r) * K + 2 * half;
    const float* __restrict__ bcol = W + (size_t)(2 * half) * M + cb + r;

    v8f acc0 = {}, acc1 = {}, acc2 = {}, acc3 = {};
    for (int k0 = 0; k0 < K; k0 += 4) {
        v2f a;
        a.x = arow[k0];
        a.y = arow[k0 + 1];
        const float* __restrict__ b = bcol + (size_t)k0 * M;
        v2f b0, b1, b2, b3;
        b0.x = b[0];  b0.y = b[M];
        b1.x = b[16]; b1.y = b[M + 16];
        b2.x = b[32]; b2.y = b[M + 32];
        b3.x = b[48]; b3.y = b[M + 48];
        acc0 = __builtin_amdgcn_wmma_f32_16x16x4_f32(false, a, false, b0, (short)0, acc0, false, false);
        acc1 = __builtin_amdgcn_wmma_f32_16x16x4_f32(false, a, false, b1, (short)0, acc1, false, false);
        acc2 = __builtin_amdgcn_wmma_f32_16x16x4_f32(false, a, false, b2, (short)0, acc2, false, false);
        acc3 = __builtin_amdgcn_wmma_f32_16x16x4_f32(false, a, false, b3, (short)0, acc3, false, false);
    }
    float* __restrict__ crow = C + (size_t)(ti * 16 + 8 * half) * M + cb + r;
    #pragma unroll
    for (int v = 0; v < 8; ++v) {
        crow[(size_t)v * M]      = acc0[v];
        crow[(size_t)v * M + 16] = acc1[v];
        crow[(size_t)v * M + 32] = acc2[v];
        crow[(size_t)v * M + 48] = acc3[v];
    }
}

// ---------------- attention coefficients: al/ar[n,h] = sum_c h[n,h,c]*a[h,c] ----------------
__global__ void attcoef_kernel(float* __restrict__ al, float* __restrict__ ar,
                               const float* __restrict__ hm,
                               const float* __restrict__ a_s, const float* __restrict__ a_d,
                               int C) {
    const int wave = (blockIdx.x * blockDim.x + threadIdx.x) >> 5;
    const int lane = threadIdx.x & 31;
    if (wave >= NN * NH) return;
    const int n = wave / NH, h = wave - n * NH;
    const float* __restrict__ hp = hm + ((size_t)n * NH + h) * C;
    float sl = 0.f, sr = 0.f;
    for (int c = lane; c < C; c += 32) {
        float v = hp[c];
        sl += v * a_s[h * C + c];
        sr += v * a_d[h * C + c];
    }
    sl = waveReduceSum(sl);
    sr = waveReduceSum(sr);
    if (lane == 0) { al[wave] = sl; ar[wave] = sr; }
}

// ---------------- edge logits + segment max over dst ----------------
__global__ void edge_logits_kernel(float* __restrict__ ebuf, unsigned* __restrict__ mbuf,
                                   const float* __restrict__ al, const float* __restrict__ ar,
                                   const int* __restrict__ src, const int* __restrict__ dst) {
    const int e = blockIdx.x * blockDim.x + threadIdx.x;
    if (e >= ETOT) return;
    int s, d;
    if (e < EE) { s = src[e]; d = dst[e]; } else { s = d = e - EE; }
    #pragma unroll
    for (int h = 0; h < NH; ++h) {
        float v = al[s * NH + h] + ar[d * NH + h];
        v = v > 0.f ? v : 0.2f * v;                 // leaky_relu(0.2)
        ebuf[(size_t)e * NH + h] = v;
        atomicMax(&mbuf[d * NH + h], fkey(v));
    }
}

// ---------------- exp(e - max) + segment sum over dst ----------------
__global__ void edge_exp_kernel(float* __restrict__ ebuf, float* __restrict__ den,
                                const unsigned* __restrict__ mbuf,
                                const int* __restrict__ dst) {
    const int e = blockIdx.x * blockDim.x + threadIdx.x;
    if (e >= ETOT) return;
    const int d = (e < EE) ? dst[e] : (e - EE);
    #pragma unroll
    for (int h = 0; h < NH; ++h) {
        float mm = funkey(mbuf[d * NH + h]);
        float ex = expf(ebuf[(size_t)e * NH + h] - mm);
        ebuf[(size_t)e * NH + h] = ex;
        unsafeAtomicAdd(&den[d * NH + h], ex);
    }
}

// ---------------- scatter: agg[d,h,:] += h[s,h,:] * alpha ----------------
// one block per edge, one wave per head
__global__ void edge_scatter_kernel(float* __restrict__ agg, const float* __restrict__ hm,
                                    const float* __restrict__ ebuf, const float* __restrict__ den,
                                    const int* __restrict__ src, const int* __restrict__ dst,
                                    int C) {
    const int e = blockIdx.x;
    const int h = threadIdx.x >> 5;
    const int lane = threadIdx.x & 31;
    int s, d;
    if (e < EE) { s = src[e]; d = dst[e]; } else { s = d = e - EE; }
    const float alpha = ebuf[(size_t)e * NH + h] / den[d * NH + h];
    const float* __restrict__ hs = hm + ((size_t)s * NH + h) * C;
    float* __restrict__ od = agg + ((size_t)d * NH + h) * C;
    for (int c = lane; c < C; c += 32) unsafeAtomicAdd(&od[c], hs[c] * alpha);
}

// ---------------- epilogue (concat layers): x = LN(ELU(agg + b)) ----------------
__global__ void node_ep_kernel(float* __restrict__ x, const float* __restrict__ agg,
                               const float* __restrict__ bias, const float* __restrict__ ls,
                               const float* __restrict__ lb, int F) {
    __shared__ float sm[256];
    const int n = blockIdx.x, tid = threadIdx.x;
    const float* __restrict__ a = agg + (size_t)n * F;
    float s = 0.f, q = 0.f;
    for (int f = tid; f < F; f += blockDim.x) {
        float v = eluf(a[f] + bias[f]);
        s += v; q += v * v;
    }
    s = blockReduceSum(s, sm);
    q = blockReduceSum(q, sm);
    const float mu = s / F;
    const float rstd = rsqrtf(q / F - mu * mu + LN_EPS);
    float* __restrict__ o = x + (size_t)n * F;
    for (int f = tid; f < F; f += blockDim.x) {
        float v = eluf(a[f] + bias[f]);
        o[f] = (v - mu) * rstd * ls[f] + lb[f];
    }
}

// ---------------- epilogue (mean-over-heads layer 2): 64 feats ----------------
__global__ void node_ep_mean_kernel(float* __restrict__ x3, const float* __restrict__ agg,
                                    const float* __restrict__ bias, const float* __restrict__ ls,
                                    const float* __restrict__ lb) {
    __shared__ float sm[64];
    const int n = blockIdx.x, c = threadIdx.x;       // 64 threads
    const float* __restrict__ a = agg + (size_t)n * 512;
    float v = 0.f;
    #pragma unroll
    for (int h = 0; h < NH; ++h) v += a[h * 64 + c];
    v = v * 0.125f + bias[c];
    v = eluf(v);
    float s = blockReduceSum(v, sm) * (1.f / 64.f);
    float d = v - s;
    float q = blockReduceSum(d * d, sm) * (1.f / 64.f);
    x3[(size_t)n * 64 + c] = d * rsqrtf(q + LN_EPS) * ls[c] + lb[c];
}

// ---------------- projection epilogue: node_out = GELU(LN(h + bp)) ----------------
__global__ void proj_ep_kernel(float* __restrict__ node_out, const float* __restrict__ hp,
                               const float* __restrict__ bp, const float* __restrict__ ls,
                               const float* __restrict__ lb) {
    __shared__ float sm[256];
    const int n = blockIdx.x, f = threadIdx.x;       // 256 threads
    float v = hp[(size_t)n * 256 + f] + bp[f];
    float mu = blockReduceSum(v, sm) * (1.f / 256.f);
    float d = v - mu;
    float var = blockReduceSum(d * d, sm) * (1.f / 256.f);
    float y = d * rsqrtf(var + LN_EPS) * ls[f] + lb[f];
    node_out[(size_t)n * 256 + f] = 0.5f * y * (1.f + erff(y * 0.70710678118654752f));
}

// ---------------- pooling logits: l[n] = node_out[n,:] . wq + bq ----------------
__global__ void logits_kernel(float* __restrict__ logits, const float* __restrict__ node_out,
                              const float* __restrict__ wq, const float* __restrict__ bq) {
    const int wave = (blockIdx.x * blockDim.x + threadIdx.x) >> 5;
    const int lane = threadIdx.x & 31;
    if (wave >= NN) return;
    const float* __restrict__ r = node_out + (size_t)wave * 256;
    float s = 0.f;
    for (int c = lane; c < 256; c += 32) s += r[c] * wq[c];
    s = waveReduceSum(s);
    if (lane == 0) logits[wave] = s + bq[0];
}

// ---------------- global softmax reduction: red[0]=max, red[1]=sum exp ----------------
__global__ void softmax_reduce_kernel(float* __restrict__ red, const float* __restrict__ logits) {
    __shared__ float sm[1024];
    const int tid = threadIdx.x;
    float m = -1e30f;
    for (int i = tid; i < NN; i += 1024) m = fmaxf(m, logits[i]);
    sm[tid] = m; __syncthreads();
    for (int s = 512; s > 0; s >>= 1) { if (tid < s) sm[tid] = fmaxf(sm[tid], sm[tid + s]); __syncthreads(); }
    const float gmax = sm[0]; __syncthreads();
    float su = 0.f;
    for (int i = tid; i < NN; i += 1024) su += expf(logits[i] - gmax);
    sm[tid] = su; __syncthreads();
    for (int s = 512; s > 0; s >>= 1) { if (tid < s) sm[tid] += sm[tid + s]; __syncthreads(); }
    if (tid == 0) { red[0] = gmax; red[1] = sm[0]; }
}

// ---------------- per-graph attention pooling ----------------
__global__ void pool_kernel(float* __restrict__ gout, const float* __restrict__ node_out,
                            const float* __restrict__ logits, const float* __restrict__ red) {
    __shared__ float sw[N_REG];
    __shared__ float sm[256];
    const int g = blockIdx.x, tid = threadIdx.x;     // 256 threads
    const int base = g * N_REG;
    const float gmax = red[0], inv = 1.f / red[1];
    for (int i = tid; i < N_REG; i += blockDim.x)
        sw[i] = expf(logits[base + i] - gmax) * inv;  // = pool_w[n]
    __syncthreads();
    float d = 0.f;
    for (int i = tid; i < N_REG; i += blockDim.x) d += sw[i];
    d = blockReduceSum(d, sm);
    d = fmaxf(d, 1e-8f);
    float num = 0.f;
    for (int i = 0; i < N_REG; ++i)
        num += node_out[(size_t)(base + i) * 256 + tid] * sw[i];
    gout[(size_t)g * 256 + tid] = num / d;
}

// ---------------- host side ----------------
extern "C" void kernel_launch(void* const* d_in, const int* in_sizes, int n_in,
                              void* d_out, int out_size, void* d_ws, size_t ws_size,
                              hipStream_t stream) {
    const float* x0  = (const float*)d_in[0];
    const int*   src = (const int*)d_in[1];
    const int*   dst = (const int*)d_in[2];
    const float* w0  = (const float*)d_in[4];
    const float* b0  = (const float*)d_in[5];
    const float* as0 = (const float*)d_in[6];
    const float* ad0 = (const float*)d_in[7];
    const float* ls0 = (const float*)d_in[8];
    const float* lb0 = (const float*)d_in[9];
    const float* w1  = (const float*)d_in[10];
    const float* b1  = (const float*)d_in[11];
    const float* as1 = (const float*)d_in[12];
    const float* ad1 = (const float*)d_in[13];
    const float* ls1 = (const float*)d_in[14];
    const float* lb1 = (const float*)d_in[15];
    const float* w2  = (const float*)d_in[16];
    const float* b2  = (const float*)d_in[17];
    const float* as2 = (const float*)d_in[18];
    const float* ad2 = (const float*)d_in[19];
    const float* ls2 = (const float*)d_in[20];
    const float* lb2 = (const float*)d_in[21];
    const float* wp  = (const float*)d_in[22];
    const float* bp  = (const float*)d_in[23];
    const float* lsp = (const float*)d_in[24];
    const float* lbp = (const float*)d_in[25];
    const float* wq  = (const float*)d_in[26];
    const float* bq  = (const float*)d_in[27];

    // workspace layout (floats)
    float*    bufH   = (float*)d_ws;                       // NN*2048 : h of current layer / proj pre-LN
    float*    bufX   = bufH + (size_t)NN * 2048;           // NN*2048 : ping
    float*    bufY   = bufX + (size_t)NN * 2048;           // NN*2048 : pong
    float*    al     = bufY + (size_t)NN * 2048;           // NN*NH
    float*    ar     = al + (size_t)NN * NH;               // NN*NH
    unsigned* mbuf   = (unsigned*)(ar + (size_t)NN * NH);  // NN*NH
    float*    den    = (float*)(mbuf + (size_t)NN * NH);   // NN*NH
    float*    ebuf   = den + (size_t)NN * NH;              // ETOT*NH
    float*    logits = ebuf + (size_t)ETOT * NH;           // NN
    float*    red    = logits + NN;                        // 2

    float* gout     = (float*)d_out;                       // (32,256)
    float* node_out = gout + (size_t)NB * 256;             // (3712,256)

    const int EBLK = (ETOT + 255) / 256;
    const int ACBLK = (NN * NH) / 8;                       // attcoef: 8 waves/block

    // ================= Layer 0: 256 -> 8x256 (concat 2048) =================
    wmma_gemm_f32<<<(232 * 32) / 8, 256, 0, stream>>>(x0, w0, bufH, NN, 256, 2048);
    attcoef_kernel<<<ACBLK, 256, 0, stream>>>(al, ar, bufH, as0, ad0, 256);
    hipMemsetAsync(mbuf, 0, (size_t)NN * NH * 4, stream);
    hipMemsetAsync(den, 0, (size_t)NN * NH * 4, stream);
    hipMemsetAsync(bufY, 0, (size_t)NN * 2048 * 4, stream);
    edge_logits_kernel<<<EBLK, 256, 0, stream>>>(ebuf, mbuf, al, ar, src, dst);
    edge_exp_kernel<<<EBLK, 256, 0, stream>>>(ebuf, den, mbuf, dst);
    edge_scatter_kernel<<<ETOT, 256, 0, stream>>>(bufY, bufH, ebuf, den, src, dst, 256);
    node_ep_kernel<<<NN, 256, 0, stream>>>(bufY, bufY, b0, ls0, lb0, 2048);

    // ================= Layer 1: 2048 -> 8x128 (concat 1024) =================
    wmma_gemm_f32<<<(232 * 16) / 8, 256, 0, stream>>>(bufY, w1, bufH, NN, 2048, 1024);
    attcoef_kernel<<<ACBLK, 256, 0, stream>>>(al, ar, bufH, as1, ad1, 128);
    hipMemsetAsync(mbuf, 0, (size_t)NN * NH * 4, stream);
    hipMemsetAsync(den, 0, (size_t)NN * NH * 4, stream);
    hipMemsetAsync(bufX, 0, (size_t)NN * 1024 * 4, stream);
    edge_logits_kernel<<<EBLK, 256, 0, stream>>>(ebuf, mbuf, al, ar, src, dst);
    edge_exp_kernel<<<EBLK, 256, 0, stream>>>(ebuf, den, mbuf, dst);
    edge_scatter_kernel<<<ETOT, 256, 0, stream>>>(bufX, bufH, ebuf, den, src, dst, 128);
    node_ep_kernel<<<NN, 256, 0, stream>>>(bufX, bufX, b1, ls1, lb1, 1024);

    // ================= Layer 2: 1024 -> 8x64 (mean -> 64) =================
    wmma_gemm_f32<<<(232 * 8) / 8, 256, 0, stream>>>(bufX, w2, bufH, NN, 1024, 512);
    attcoef_kernel<<<ACBLK, 256, 0, stream>>>(al, ar, bufH, as2, ad2, 64);
    hipMemsetAsync(mbuf, 0, (size_t)NN * NH * 4, stream);
    hipMemsetAsync(den, 0, (size_t)NN * NH * 4, stream);
    hipMemsetAsync(bufY, 0, (size_t)NN * 512 * 4, stream);
    edge_logits_kernel<<<EBLK, 256, 0, stream>>>(ebuf, mbuf, al, ar, src, dst);
    edge_exp_kernel<<<EBLK, 256, 0, stream>>>(ebuf, den, mbuf, dst);
    edge_scatter_kernel<<<ETOT, 256, 0, stream>>>(bufY, bufH, ebuf, den, src, dst, 64);
    node_ep_mean_kernel<<<NN, 64, 0, stream>>>(bufX, bufY, b2, ls2, lb2);   // x3 (NN,64) in bufX

    // ================= Projection 64 -> 256, LN + GELU =================
    wmma_gemm_f32<<<(232 * 4) / 8, 256, 0, stream>>>(bufX, wp, bufH, NN, 64, 256);
    proj_ep_kernel<<<NN, 256, 0, stream>>>(node_out, bufH, bp, lsp, lbp);

    // ================= Global attention pooling =================
    logits_kernel<<<NN / 8, 256, 0, stream>>>(logits, node_out, wq, bq);
    softmax_reduce_kernel<<<1, 1024, 0, stream>>>(red, logits);
    pool_kernel<<<NB, 256, 0, stream>>>(gout, node_out, logits, red);
}